// Solver_73263552135357
// MI455X (gfx1250) — compile-verified
//
#include <hip/hip_runtime.h>

// ---------------------------------------------------------------------------
// MI455X (gfx1250) solver: per-pixel residual = J^T H^{-1} J, N=31.
//   rate = S x;  J = -(S^T (t*(k/rate - 1) + 1/rate)) - 1/x
//   w = (t*k + 1)/rate^2;  H = S diag(w) S^T + diag(1/x^2)
//   residual = || L^{-1} J ||^2  with  H = L L^T   (no backward solve needed)
//
// One wave (32 lanes) per pixel. 32x32x32 f32 Gram on V_WMMA_F32_16X16X4_F32
// (32 wmma / pixel). Cholesky + forward solve are row-per-lane. Broadcast
// strategy balances the two issue pipes:
//   - bulk (rank-1 update, matvec) broadcasts: ds_swizzle_b32 with immediate
//     BCAST32 pattern (offset = lane<<5)  -> DS pipe, zero VALU cost
//   - serial-critical-path broadcasts (pivot, solve chain): v_readlane
// Compute bound: ~2.6 GFLOP total vs ~16 MB HBM traffic (<1us @ 23.3 TB/s).
// ---------------------------------------------------------------------------

typedef float v2f __attribute__((ext_vector_type(2)));
typedef float v8f __attribute__((ext_vector_type(8)));

#define N_FR 31          // frames (matrix order)
#define NP   32          // padded order
#define LDSP 33          // padded LDS row stride (conflict-free: gcd(33,64)=1)
#define WAVES_PER_BLOCK 8
#define PPW  8           // pixels per wave
#define NPIX (256 * 256)

// Low-latency broadcast via v_readlane (VALU, lands in SGPR).
__device__ __forceinline__ float bcastf(float v, int l) {
    return __int_as_float(__builtin_amdgcn_readlane(__float_as_int(v), l));
}

// DS-pipe broadcast of lane L via ds_swizzle_b32 (group-of-32: or_mask = L).
// Immediate-encoded: no index VGPR, no LDS memory access.
template <int L>
__device__ __forceinline__ float swzbcast(float v) {
    return __int_as_float(
        __builtin_amdgcn_ds_swizzle(__float_as_int(v), (L << 5)));
}

// acc += sum_{A<NMAX} Sbase[A*stride] * bcast(vec, A)   (swizzle broadcasts)
template <int A, int NMAX>
struct MatvecAcc {
    static __device__ __forceinline__ float run(const float* Sbase, int stride,
                                                float vec, float acc) {
        acc = fmaf(Sbase[A * stride], swzbcast<A>(vec), acc);
        return MatvecAcc<A + 1, NMAX>::run(Sbase, stride, vec, acc);
    }
};
template <int NMAX>
struct MatvecAcc<NMAX, NMAX> {
    static __device__ __forceinline__ float run(const float*, int, float,
                                                float acc) { return acc; }
};

// Cholesky trailing rank-1 update: h[E] -= lij * L[E][j] for E in (J, NP)
template <int J, int E>
struct CholUpdate {
    static __device__ __forceinline__ void run(float (&h)[NP], float lij) {
        float lej = swzbcast<E>(lij);        // L[E][J], E > J -> valid lanes
        h[E] = fmaf(-lij, lej, h[E]);
        CholUpdate<J, E + 1>::run(h, lij);
    }
};
template <int J>
struct CholUpdate<J, NP> {
    static __device__ __forceinline__ void run(float (&)[NP], float) {}
};

// One right-looking Cholesky column step (J), then recurse.
template <int J>
struct CholStep {
    static __device__ __forceinline__ void run(float (&h)[NP], int lane) {
        float hjj = bcastf(h[J], J);                 // pivot (critical path)
        float isj = __builtin_amdgcn_rsqf(hjj);      // 1/sqrt
        float lij = h[J] * isj;                      // L[i][J] (i >= J valid)
        h[J] = (lane == J) ? isj : lij;              // stash 1/L[J][J] on diag
        CholUpdate<J, J + 1>::run(h, lij);
        CholStep<J + 1>::run(h, lane);
    }
};
template <>
struct CholStep<NP> {
    static __device__ __forceinline__ void run(float (&)[NP], int) {}
};

__global__ __launch_bounds__(256) void solver_newton_residual(
    const float* __restrict__ tp,   // time_points, (HW, 31)
    const float* __restrict__ px,   // pixels,      (HW, 31)
    const float* __restrict__ Sg,   // S, (31, 31)
    const float* __restrict__ tscal,
    float* __restrict__ out)        // (HW)
{
    __shared__ float Spad[NP * LDSP];                    // zero-padded S
    __shared__ float Hw[WAVES_PER_BLOCK][NP * LDSP];     // per-wave H tile

    const int tid  = threadIdx.x;
    const int lane = tid & 31;
    const int wv   = tid >> 5;
    const float t  = tscal[0];

    // Stage S into padded LDS (rows/cols >= 31 are zero).
    for (int idx = tid; idx < NP * LDSP; idx += 256) {
        int r = idx / LDSP, c = idx % LDSP;
        float v = 0.0f;
        if (r < N_FR && c < N_FR) v = Sg[r * N_FR + c];
        Spad[idx] = v;
    }
    __syncthreads();

    // S in WMMA A-layout registers. For V_WMMA_F32_16X16X4_F32 both the A
    // operand (M across lanes 0..15, K = reg + 2*(lane>=16)) and the B
    // operand (N across lanes 0..15, K = reg + 2*(lane>=16)) use the SAME
    // (lane,reg) -> (row,K) mapping, so SA[t] is the A tile for M-rows
    // 16t..16t+15 AND (scaled by w[K]) the B tile for N-cols 16t..16t+15:
    //   B[K][N] = w[K] * S[N][K]   which is exactly what the Gram needs.
    float SA[2][16];
    {
        const int mrow  = lane & 15;
        const int khalf = 2 * (lane >> 4);
        #pragma unroll
        for (int t2 = 0; t2 < 2; ++t2)
            #pragma unroll
            for (int g = 0; g < 8; ++g)
                #pragma unroll
                for (int r = 0; r < 2; ++r)
                    SA[t2][2 * g + r] =
                        Spad[(mrow + 16 * t2) * LDSP + (4 * g + r + khalf)];
    }

    const int gwave = blockIdx.x * WAVES_PER_BLOCK + wv;
    float* Hl = &Hw[wv][0];

    #pragma unroll 1
    for (int it = 0; it < PPW; ++it) {
        const int p    = gwave * PPW + it;
        const int base = p * N_FR;

        // lane a holds frame a (pad: x=1, k=0)
        float xv = 1.0f, kv = 0.0f;
        if (lane < N_FR) { xv = tp[base + lane]; kv = px[base + lane]; }

        // rate = S x   (row i of S from LDS, x broadcast on the DS pipe)
        float rate = MatvecAcc<0, N_FR>::run(&Spad[lane * LDSP], 1, xv, 0.0f);

        const float invr = 1.0f / rate;   // lane-parallel IEEE divide
        const float invx = 1.0f / xv;

        // m = t*(k/rate - 1) + 1/rate ; w = (t*k+1)/rate^2 ; pad lanes -> 0
        float m   = fmaf(t, fmaf(kv, invr, -1.0f), invr);
        float wgt = fmaf(t, kv, 1.0f) * invr * invr;
        if (lane >= N_FR) { m = 0.0f; wgt = 0.0f; }

        // J = -(S^T m) - 1/x   (column i of S from LDS: true transpose)
        float sm   = MatvecAcc<0, N_FR>::run(&Spad[lane], LDSP, m, 0.0f);
        float Jv   = -sm - invx;
        float diag = invx * invx;
        if (lane >= N_FR) { Jv = 0.0f; diag = 1.0f; }

        // Scaled B tiles: Bs[t][2g+r] = SA[t][2g+r] * w[K], K = 4g+r+khalf
        float Bs[2][16];
        #pragma unroll
        for (int g = 0; g < 8; ++g)
            #pragma unroll
            for (int r = 0; r < 2; ++r) {
                float wlo = bcastf(wgt, 4 * g + r);
                float whi = bcastf(wgt, 4 * g + r + 2);
                float ws  = (lane < 16) ? wlo : whi;
                Bs[0][2 * g + r] = SA[0][2 * g + r] * ws;
                Bs[1][2 * g + r] = SA[1][2 * g + r] * ws;
            }

        // H = A * B : 2x2 tiles of 16x16, K=32 in 8 chained wmma per tile.
        v8f C00 = {}; v8f C01 = {}; v8f C10 = {}; v8f C11 = {};
        #pragma unroll
        for (int g = 0; g < 8; ++g) {
            v2f a0 = { SA[0][2 * g], SA[0][2 * g + 1] };
            v2f a1 = { SA[1][2 * g], SA[1][2 * g + 1] };
            v2f b0 = { Bs[0][2 * g], Bs[0][2 * g + 1] };
            v2f b1 = { Bs[1][2 * g], Bs[1][2 * g + 1] };
            C00 = __builtin_amdgcn_wmma_f32_16x16x4_f32(false, a0, false, b0,
                                                        (short)0, C00, false, false);
            C01 = __builtin_amdgcn_wmma_f32_16x16x4_f32(false, a0, false, b1,
                                                        (short)0, C01, false, false);
            C10 = __builtin_amdgcn_wmma_f32_16x16x4_f32(false, a1, false, b0,
                                                        (short)0, C10, false, false);
            C11 = __builtin_amdgcn_wmma_f32_16x16x4_f32(false, a1, false, b1,
                                                        (short)0, C11, false, false);
        }

        // Relayout C accumulators -> LDS rows (C layout: VGPR r holds
        // M = 16*mT + r + 8*(lane>=16), N = 16*nT + lane%16).
        {
            const int rowoff = 8 * (lane >> 4);
            const int col    = lane & 15;
            #pragma unroll
            for (int r = 0; r < 8; ++r) {
                Hl[(r + rowoff) * LDSP + col]           = C00[r];
                Hl[(r + rowoff) * LDSP + col + 16]      = C01[r];
                Hl[(16 + r + rowoff) * LDSP + col]      = C10[r];
                Hl[(16 + r + rowoff) * LDSP + col + 16] = C11[r];
            }
        }
        __syncthreads();

        // diag(1/x^2) (+1 on the pad row): lane i RMWs H[i][i]
        Hl[lane * LDSP + lane] += diag;

        // lane i pulls row i into registers
        float h[NP];
        #pragma unroll
        for (int a = 0; a < NP; ++a) h[a] = Hl[lane * LDSP + a];

        // Right-looking Cholesky, row-per-lane; diagonal slots hold 1/L[j][j].
        CholStep<0>::run(h, lane);

        // Forward solve L y = J ;  residual = sum y_j^2 (uniform in wave).
        // Lane j's acc is last read at step j, so the inverse-diagonal stash
        // in h[j] is safe for the rank-1 acc updates (i > j only).
        float acc = Jv;
        float res = 0.0f;
        #pragma unroll
        for (int j = 0; j < NP; ++j) {
            float accj = bcastf(acc, j);
            float dinv = bcastf(h[j], j);        // 1/L[j][j]
            float yj   = accj * dinv;
            res = fmaf(yj, yj, res);
            acc = fmaf(-h[j], yj, acc);
        }

        if (lane == 0) out[p] = res;
        __syncthreads();   // protect Hl reuse next iteration
    }
}

extern "C" void kernel_launch(void* const* d_in, const int* in_sizes, int n_in,
                              void* d_out, int out_size, void* d_ws, size_t ws_size,
                              hipStream_t stream) {
    (void)in_sizes; (void)n_in; (void)d_ws; (void)ws_size; (void)out_size;
    const float* tp = (const float*)d_in[0];   // time_points (65536*31)
    const float* px = (const float*)d_in[1];   // pixels      (65536*31)
    const float* S  = (const float*)d_in[2];   // S (31*31)
    const float* t  = (const float*)d_in[3];   // scalar t
    float* out = (float*)d_out;                // (65536)

    const int pixels_per_block = WAVES_PER_BLOCK * PPW;     // 64
    const int blocks = NPIX / pixels_per_block;             // 1024
    solver_newton_residual<<<blocks, 256, 0, stream>>>(tp, px, S, t, out);
}